// Phi2_Network_13434657702173
// MI455X (gfx1250) — compile-verified
//
#include <hip/hip_runtime.h>

typedef __attribute__((ext_vector_type(2))) float v2f;
typedef __attribute__((ext_vector_type(8))) float v8f;

#define NCOL 300
#define NTILE 19                    // ceil(300/16) column tiles
#define BFRAG_OFF 0
#define BFRAG_FLOATS (NTILE * 32 * 2)   // 1216 floats: B-fragments in lane layout
#define W32_OFF BFRAG_FLOATS
#define W32_PAD (NTILE * 16)            // 304 (zero padded)
#define W31_OFF (W32_OFF + W32_PAD)     // 3 floats

// ---------------------------------------------------------------------------
// Prep: fold diagonals into weights and pre-pack the WMMA B-fragments.
//   W21[j][k]  = varphi1[j][k] * (l1[k] - l2[j])          (k<3; k==3 pad = 0)
//   W32[j]     = varphi2[j] * (l2[j] - l3)
//   W31[k]     = -sum_j W32[j] * varphi1[j][k]
// B-fragment layout for V_WMMA_F32_16X16X4_F32 (B is 4x16, ISA 32-bit layout):
//   lane L: N = L%16 ; VGPR0 holds K=(L<16?0:2), VGPR1 holds K=(L<16?1:3)
// ---------------------------------------------------------------------------
__global__ __launch_bounds__(256) void phi2_prep(
    const float* __restrict__ varphi1,   // [300,3]
    const float* __restrict__ varphi2,   // [300]
    const float* __restrict__ l1,        // [3]
    const float* __restrict__ l2,        // [300]
    const float* __restrict__ l3,        // [1]
    float* __restrict__ ws)
{
    const int tid = threadIdx.x;
    const float l3v = l3[0];

    // padded W32
    for (int j = tid; j < W32_PAD; j += 256) {
        ws[W32_OFF + j] = (j < NCOL) ? varphi2[j] * (l2[j] - l3v) : 0.0f;
    }

    // B-fragments: i = tile*32 + lane
    for (int i = tid; i < NTILE * 32; i += 256) {
        const int t = i >> 5;
        const int l = i & 31;
        const int j = t * 16 + (l & 15);    // output column (N)
        float v0 = 0.0f, v1 = 0.0f;
        if (j < NCOL) {
            const float sj = -l2[j];
            if (l < 16) {                    // K = 0,1
                v0 = varphi1[j * 3 + 0] * (l1[0] + sj);
                v1 = varphi1[j * 3 + 1] * (l1[1] + sj);
            } else {                         // K = 2 (bias row), K = 3 pad
                v0 = varphi1[j * 3 + 2] * (l1[2] + sj);
                v1 = 0.0f;
            }
        }
        ws[BFRAG_OFF + i * 2 + 0] = v0;
        ws[BFRAG_OFF + i * 2 + 1] = v1;
    }

    // W31 (3 dot products of length 300 — negligible)
    if (tid < 3) {
        const int k = tid;
        float s = 0.0f;
        for (int j = 0; j < NCOL; ++j) {
            const float w32j = varphi2[j] * (l2[j] - l3v);
            s -= w32j * varphi1[j * 3 + k];
        }
        ws[W31_OFF + k] = s;
    }
}

// Single-instruction relu: the compiler re-canonicalizes every source-level
// fmax/fmed3 formulation into a 2-op canonicalize+max pair because it cannot
// prove the WMMA result is canonical. Force the exact hardware op instead.
// (The F32 WMMA has no WMMA->VALU NOP requirement per the ISA hazard table,
// confirmed by the compiler emitting direct VALU consumers of the D regs.)
__device__ __forceinline__ float relu1(float v) {
    float r;
    asm("v_max_num_f32 %0, 0, %1" : "=v"(r) : "v"(v));
    return r;
}

// ---------------------------------------------------------------------------
// Main: one wave per 16 rows. A-fragment (16x4 fp32, ISA layout):
//   lanes 0-15 : M=L, VGPR0=K0=x0, VGPR1=K1=x1
//   lanes 16-31: M=L-16, VGPR0=K2=1 (bias), VGPR1=K3=0 (pad)
// For each of 19 column tiles: d = wmma_f32_16x16x4_f32(a, b, 0), then fused
// epilogue acc[r] += relu(d[r]) * w32[col]. D layout: VGPR r, lanes 0-15 hold
// row r / cols 0-15, lanes 16-31 hold row r+8. Final shfl_xor reduction over
// each 16-lane half yields the per-row sum; add the W31*xa skip connection.
// ---------------------------------------------------------------------------
__global__ __launch_bounds__(256) void phi2_main(
    const float* __restrict__ x,     // [B,2]
    const float* __restrict__ ws,
    float* __restrict__ y,           // [B]
    int nrows)
{
    const int lane = threadIdx.x & 31;
    const int wave = threadIdx.x >> 5;
    const long long rowBase = ((long long)blockIdx.x * 8 + wave) * 16;
    if (rowBase >= nrows) return;    // wave-uniform exit; EXEC stays full

    // ---- A fragment ----
    v2f a;
    {
        const int m = lane & 15;
        long long row = rowBase + m;
        if (row > nrows - 1) row = nrows - 1;   // safety clamp (B%16==0 anyway)
        if (lane < 16) {
            a.x = x[row * 2 + 0];
            a.y = x[row * 2 + 1];
        } else {
            a.x = 1.0f;   // bias column (K=2)
            a.y = 0.0f;   // pad       (K=3)
        }
    }

    const float* __restrict__ bf  = ws + BFRAG_OFF;
    const float* __restrict__ w32 = ws + W32_OFF;
    const float* __restrict__ w31 = ws + W31_OFF;

    v8f acc = {0.f, 0.f, 0.f, 0.f, 0.f, 0.f, 0.f, 0.f};

    for (int t = 0; t < NTILE; ++t) {
        const v2f b = *(const v2f*)(bf + (t * 32 + lane) * 2);
        v8f c = {};
        // D = A(16x4) x B(4x16) + 0, full fp32
        v8f d = __builtin_amdgcn_wmma_f32_16x16x4_f32(
            /*neg_a=*/false, a, /*neg_b=*/false, b,
            /*c_mod=*/(short)0, c, /*reuse_a=*/false, /*reuse_b=*/false);
        const float wv = w32[t * 16 + (lane & 15)];
#pragma unroll
        for (int r = 0; r < 8; ++r) {
            acc[r] += relu1(d[r]) * wv;       // v_max_num + v_fmac per element
        }
    }

    // Reduce over the 16 columns of each tile-half (xor masks 1..8 never
    // cross the 16-lane boundary, so halves stay independent).
#pragma unroll
    for (int r = 0; r < 8; ++r) {
        float s = acc[r];
        s += __shfl_xor(s, 1, 32);
        s += __shfl_xor(s, 2, 32);
        s += __shfl_xor(s, 4, 32);
        s += __shfl_xor(s, 8, 32);
        acc[r] = s;
    }

    // lane 0 writes rows rowBase+0..7, lane 16 writes rows rowBase+8..15
    if ((lane & 15) == 0) {
        const float c0 = w31[0], c1 = w31[1], c2 = w31[2];
        const long long base = rowBase + ((lane >> 4) << 3);
#pragma unroll
        for (int r = 0; r < 8; ++r) {
            const long long orow = base + r;
            if (orow < nrows) {
                const float x0 = x[orow * 2 + 0];
                const float x1 = x[orow * 2 + 1];
                y[orow] = acc[r] + c0 * x0 + c1 * x1 + c2;
            }
        }
    }
}

// ---------------------------------------------------------------------------
extern "C" void kernel_launch(void* const* d_in, const int* in_sizes, int n_in,
                              void* d_out, int out_size, void* d_ws, size_t ws_size,
                              hipStream_t stream) {
    const float* x       = (const float*)d_in[0];
    const float* varphi1 = (const float*)d_in[1];
    const float* varphi2 = (const float*)d_in[2];
    const float* l1      = (const float*)d_in[3];
    const float* l2      = (const float*)d_in[4];
    const float* l3      = (const float*)d_in[5];
    float* y  = (float*)d_out;
    float* ws = (float*)d_ws;

    const int B = in_sizes[0] / 2;           // x is [B,2]

    phi2_prep<<<1, 256, 0, stream>>>(varphi1, varphi2, l1, l2, l3, ws);

    const int wavesNeeded = (B + 15) / 16;   // one wave per 16 rows
    const int grid = (wavesNeeded + 7) / 8;  // 8 waves (256 threads) per block
    phi2_main<<<grid, 256, 0, stream>>>(x, ws, y, B);
}